// PlainTensorProduct_10144712753190
// MI455X (gfx1250) — compile-verified
//
#include <hip/hip_runtime.h>

// Cartesian tensor product, B=2048, F=128, L_MAX=3 (23 paths, 247*F outputs/row).
// Memory-bound (~343 MB @ 23.3 TB/s => ~15us floor). Strategy:
//   - one block (128 thr = 4 wave32) per batch row; thread t <-> feature t
//   - stage the 80 input component-vectors (40 KB) to LDS with CDNA5
//     global_load_async_to_lds_b128 (ASYNCcnt), s_wait_asynccnt + barrier
//   - fully unrolled f32 VALU contractions + sym/detrace, coalesced b32 stores

#define FDIM   128
#define NCOMP  40          // 1+3+9+27 components per (x or y) set
#define OUTC   247         // output components per row

// ---------- CDNA5 async global -> LDS staging ----------
__device__ __forceinline__ void async_ld_b128(unsigned lds_byte, const float* g) {
    unsigned long long ga = (unsigned long long)g;
    asm volatile("global_load_async_to_lds_b128 %0, %1, off"
                 :: "v"(lds_byte), "v"(ga)
                 : "memory");
}

__device__ __forceinline__ void wait_async0() {
    asm volatile("s_wait_asynccnt 0x0" ::: "memory");
}

__device__ __forceinline__ void stage(const float* __restrict__ g, unsigned ldsb,
                                      int nfloats, int t) {
    const int nchunk = nfloats >> 2;   // 16-byte chunks
    for (int c = t; c < nchunk; c += FDIM)
        async_ld_b128(ldsb + (unsigned)c * 16u, g + 4 * c);
}

// ---------- symmetrize + detrace helpers (store directly) ----------
__device__ __forceinline__ void symdet2_store(const float tm[9], float nrm,
                                              float* __restrict__ O, int base) {
    const float tr3 = (tm[0] + tm[4] + tm[8]) * (1.0f / 3.0f);
    #pragma unroll
    for (int a = 0; a < 3; ++a)
        #pragma unroll
        for (int c = 0; c < 3; ++c) {
            float s = 0.5f * (tm[3 * a + c] + tm[3 * c + a]);
            if (a == c) s -= tr3;
            O[(base + 3 * a + c) * FDIM] = nrm * s;
        }
}

__device__ __forceinline__ void symdet3_store(const float tm[27], float nrm,
                                              float* __restrict__ O, int base) {
    float ts[27];
    #pragma unroll
    for (int a = 0; a < 3; ++a)
        #pragma unroll
        for (int c = 0; c < 3; ++c)
            #pragma unroll
            for (int d = 0; d < 3; ++d)
                ts[9*a + 3*c + d] =
                    (tm[9*a + 3*c + d] + tm[9*a + 3*d + c] + tm[9*c + 3*a + d] +
                     tm[9*c + 3*d + a] + tm[9*d + 3*a + c] + tm[9*d + 3*c + a]) *
                    (1.0f / 6.0f);
    float v[3];
    #pragma unroll
    for (int a = 0; a < 3; ++a)
        v[a] = ts[9*a + 0] + ts[9*a + 3*1 + 1] + ts[9*a + 3*2 + 2];
    #pragma unroll
    for (int a = 0; a < 3; ++a)
        #pragma unroll
        for (int c = 0; c < 3; ++c)
            #pragma unroll
            for (int d = 0; d < 3; ++d) {
                float corr = 0.0f;
                if (a == c) corr += v[d];
                if (a == d) corr += v[c];
                if (c == d) corr += v[a];
                O[(base + 9*a + 3*c + d) * FDIM] =
                    nrm * (ts[9*a + 3*c + d] - 0.2f * corr);
            }
}

// ---------- LDS accessors: X at comps 0..39, Y at comps 40..79 ----------
#define SX0        sh[(0) * FDIM + t]
#define SX1(a)     sh[(1 + (a)) * FDIM + t]
#define SX2(i,j)   sh[(4 + 3*(i) + (j)) * FDIM + t]
#define SX3(i,j,k) sh[(13 + 9*(i) + 3*(j) + (k)) * FDIM + t]
#define SY0        sh[(40) * FDIM + t]
#define SY1(a)     sh[(41 + (a)) * FDIM + t]
#define SY2(i,j)   sh[(44 + 3*(i) + (j)) * FDIM + t]
#define SY3(i,j,k) sh[(53 + 9*(i) + 3*(j) + (k)) * FDIM + t]
#define ST(c, v)   O[(c) * FDIM] = (v)

__global__ __launch_bounds__(FDIM) void cart_tp_kernel(
    const float* __restrict__ x0, const float* __restrict__ y0,
    const float* __restrict__ x1, const float* __restrict__ y1,
    const float* __restrict__ x2, const float* __restrict__ y2,
    const float* __restrict__ x3, const float* __restrict__ y3,
    float* __restrict__ out)
{
    __shared__ float sh[2 * NCOMP * FDIM];   // 40 KB
    const int t = threadIdx.x;
    const long long b = blockIdx.x;

    // low 32 bits of a generic LDS pointer == LDS byte offset
    const unsigned shb = (unsigned)(unsigned long long)(const void*)sh;

    stage(x0 + b * (1  * FDIM), shb + 0u  * FDIM * 4u, 1  * FDIM, t);
    stage(x1 + b * (3  * FDIM), shb + 1u  * FDIM * 4u, 3  * FDIM, t);
    stage(x2 + b * (9  * FDIM), shb + 4u  * FDIM * 4u, 9  * FDIM, t);
    stage(x3 + b * (27 * FDIM), shb + 13u * FDIM * 4u, 27 * FDIM, t);
    stage(y0 + b * (1  * FDIM), shb + 40u * FDIM * 4u, 1  * FDIM, t);
    stage(y1 + b * (3  * FDIM), shb + 41u * FDIM * 4u, 3  * FDIM, t);
    stage(y2 + b * (9  * FDIM), shb + 44u * FDIM * 4u, 9  * FDIM, t);
    stage(y3 + b * (27 * FDIM), shb + 53u * FDIM * 4u, 27 * FDIM, t);

    wait_async0();      // per-wave: all my async LDS writes landed
    __syncthreads();    // cross-wave visibility of staged tiles

    float* O = out + b * (long long)(OUTC * FDIM) + t;

    // ================= l3 = 0 : offsets 0..3 =================
    {
        ST(0, SX0 * SY0);
        float s1 = 0.0f;
        #pragma unroll
        for (int a = 0; a < 3; ++a) s1 += SX1(a) * SY1(a);
        ST(1, s1);
        float s2 = 0.0f;
        #pragma unroll
        for (int i = 0; i < 3; ++i)
            #pragma unroll
            for (int j = 0; j < 3; ++j) s2 += SX2(i, j) * SY2(i, j);
        ST(2, (2.0f / 3.0f) * s2);
        float s3 = 0.0f;
        #pragma unroll
        for (int i = 0; i < 3; ++i)
            #pragma unroll
            for (int j = 0; j < 3; ++j)
                #pragma unroll
                for (int k = 0; k < 3; ++k) s3 += SX3(i, j, k) * SY3(i, j, k);
        ST(3, 0.4f * s3);
    }

    // ================= l3 = 1 : offsets 4,7,10,13,16,19 =================
    #pragma unroll
    for (int a = 0; a < 3; ++a) ST(4 + a, SX0 * SY1(a));          // (0,1)
    #pragma unroll
    for (int a = 0; a < 3; ++a) ST(7 + a, SX1(a) * SY0);          // (1,0)
    #pragma unroll
    for (int a = 0; a < 3; ++a) {                                 // (1,2)
        float s = 0.0f;
        #pragma unroll
        for (int i = 0; i < 3; ++i) s += SX1(i) * SY2(i, a);
        ST(10 + a, s);
    }
    #pragma unroll
    for (int a = 0; a < 3; ++a) {                                 // (2,1)
        float s = 0.0f;
        #pragma unroll
        for (int j = 0; j < 3; ++j) s += SX2(a, j) * SY1(j);
        ST(13 + a, s);
    }
    #pragma unroll
    for (int c = 0; c < 3; ++c) {                                 // (2,3) norm 2/3
        float s = 0.0f;
        #pragma unroll
        for (int i = 0; i < 3; ++i)
            #pragma unroll
            for (int j = 0; j < 3; ++j) s += SX2(i, j) * SY3(i, j, c);
        ST(16 + c, (2.0f / 3.0f) * s);
    }
    #pragma unroll
    for (int i = 0; i < 3; ++i) {                                 // (3,2) norm 2/3
        float s = 0.0f;
        #pragma unroll
        for (int j = 0; j < 3; ++j)
            #pragma unroll
            for (int k = 0; k < 3; ++k) s += SX3(i, j, k) * SY2(j, k);
        ST(19 + i, (2.0f / 3.0f) * s);
    }

    // ===== l3 = 2 : offsets 22,31,40,49,58,67,76 =====
    {
        float tm[9];
        #pragma unroll
        for (int a = 0; a < 3; ++a)                               // (0,2)
            #pragma unroll
            for (int c = 0; c < 3; ++c) tm[3*a + c] = SX0 * SY2(a, c);
        symdet2_store(tm, 1.0f, O, 22);

        #pragma unroll
        for (int a = 0; a < 3; ++a)                               // (1,1) norm 3/4
            #pragma unroll
            for (int c = 0; c < 3; ++c) tm[3*a + c] = SX1(a) * SY1(c);
        symdet2_store(tm, 0.75f, O, 31);

        #pragma unroll
        for (int j = 0; j < 3; ++j)                               // (1,3)
            #pragma unroll
            for (int k = 0; k < 3; ++k) {
                float s = 0.0f;
                #pragma unroll
                for (int i = 0; i < 3; ++i) s += SX1(i) * SY3(i, j, k);
                tm[3*j + k] = s;
            }
        symdet2_store(tm, 1.0f, O, 40);

        #pragma unroll
        for (int a = 0; a < 3; ++a)                               // (2,0)
            #pragma unroll
            for (int c = 0; c < 3; ++c) tm[3*a + c] = SX2(a, c) * SY0;
        symdet2_store(tm, 1.0f, O, 49);

        #pragma unroll
        for (int i = 0; i < 3; ++i)                               // (2,2)
            #pragma unroll
            for (int k = 0; k < 3; ++k) {
                float s = 0.0f;
                #pragma unroll
                for (int j = 0; j < 3; ++j) s += SX2(i, j) * SY2(j, k);
                tm[3*i + k] = s;
            }
        symdet2_store(tm, 1.0f, O, 58);

        #pragma unroll
        for (int i = 0; i < 3; ++i)                               // (3,1)
            #pragma unroll
            for (int j = 0; j < 3; ++j) {
                float s = 0.0f;
                #pragma unroll
                for (int k = 0; k < 3; ++k) s += SX3(i, j, k) * SY1(k);
                tm[3*i + j] = s;
            }
        symdet2_store(tm, 1.0f, O, 67);

        #pragma unroll
        for (int i = 0; i < 3; ++i)                               // (3,3) norm 3/4
            #pragma unroll
            for (int m = 0; m < 3; ++m) {
                float s = 0.0f;
                #pragma unroll
                for (int j = 0; j < 3; ++j)
                    #pragma unroll
                    for (int k = 0; k < 3; ++k) s += SX3(i, j, k) * SY3(j, k, m);
                tm[3*i + m] = s;
            }
        symdet2_store(tm, 0.75f, O, 76);
    }

    // ===== l3 = 3 : offsets 85,112,139,166,193,220 =====
    {
        float tm[27];
        #pragma unroll
        for (int a = 0; a < 3; ++a)                               // (0,3)
            #pragma unroll
            for (int c = 0; c < 3; ++c)
                #pragma unroll
                for (int d = 0; d < 3; ++d) tm[9*a + 3*c + d] = SX0 * SY3(a, c, d);
        symdet3_store(tm, 1.0f, O, 85);

        #pragma unroll
        for (int i = 0; i < 3; ++i)                               // (1,2) norm 5/9
            #pragma unroll
            for (int j = 0; j < 3; ++j)
                #pragma unroll
                for (int k = 0; k < 3; ++k) tm[9*i + 3*j + k] = SX1(i) * SY2(j, k);
        symdet3_store(tm, 5.0f / 9.0f, O, 112);

        #pragma unroll
        for (int i = 0; i < 3; ++i)                               // (2,1) norm 5/9
            #pragma unroll
            for (int j = 0; j < 3; ++j)
                #pragma unroll
                for (int k = 0; k < 3; ++k) tm[9*i + 3*j + k] = SX2(i, j) * SY1(k);
        symdet3_store(tm, 5.0f / 9.0f, O, 139);

        #pragma unroll
        for (int i = 0; i < 3; ++i)                               // (2,3) norm 5/6
            #pragma unroll
            for (int k = 0; k < 3; ++k)
                #pragma unroll
                for (int m = 0; m < 3; ++m) {
                    float s = 0.0f;
                    #pragma unroll
                    for (int j = 0; j < 3; ++j) s += SX2(i, j) * SY3(j, k, m);
                    tm[9*i + 3*k + m] = s;
                }
        symdet3_store(tm, 5.0f / 6.0f, O, 166);

        #pragma unroll
        for (int a = 0; a < 3; ++a)                               // (3,0)
            #pragma unroll
            for (int c = 0; c < 3; ++c)
                #pragma unroll
                for (int d = 0; d < 3; ++d) tm[9*a + 3*c + d] = SX3(a, c, d) * SY0;
        symdet3_store(tm, 1.0f, O, 193);

        #pragma unroll
        for (int i = 0; i < 3; ++i)                               // (3,2) norm 5/6
            #pragma unroll
            for (int j = 0; j < 3; ++j)
                #pragma unroll
                for (int m = 0; m < 3; ++m) {
                    float s = 0.0f;
                    #pragma unroll
                    for (int k = 0; k < 3; ++k) s += SX3(i, j, k) * SY2(k, m);
                    tm[9*i + 3*j + m] = s;
                }
        symdet3_store(tm, 5.0f / 6.0f, O, 220);
    }
}

extern "C" void kernel_launch(void* const* d_in, const int* in_sizes, int n_in,
                              void* d_out, int out_size, void* d_ws, size_t ws_size,
                              hipStream_t stream) {
    // setup_inputs() dict order is interleaved: x0,y0,x1,y1,x2,y2,x3,y3
    const float* x0 = (const float*)d_in[0];
    const float* y0 = (const float*)d_in[1];
    const float* x1 = (const float*)d_in[2];
    const float* y1 = (const float*)d_in[3];
    const float* x2 = (const float*)d_in[4];
    const float* y2 = (const float*)d_in[5];
    const float* x3 = (const float*)d_in[6];
    const float* y3 = (const float*)d_in[7];
    float* out = (float*)d_out;

    const int B = in_sizes[0] / FDIM;   // x0 is (B, F)
    cart_tp_kernel<<<B, FDIM, 0, stream>>>(x0, y0, x1, y1, x2, y2, x3, y3, out);

    (void)n_in; (void)out_size; (void)d_ws; (void)ws_size;
}